// Apply_Mask_36009005810157
// MI455X (gfx1250) — compile-verified
//
#include <hip/hip_runtime.h>
#include <stdint.h>

// DropBlock-style mask: per 32x32 slice, find argmax, zero a clipped
// (2*half+1)^2 box around it, rescale survivors by 1024/sum(S); gated by T.
// Pure bandwidth kernel (~256 MB @ 23.3 TB/s ~ 11us). CDNA5 path used:
// async global->LDS staging (ASYNCcnt) with per-wave double buffering,
// wave32 shuffle argmax reduction, B128 loads/stores throughout.

#define WAVES_PER_BLOCK 4
#define SLICES_PER_WAVE 4
#define HH 32
#define WW 32
#define SLICE_ELEMS (HH * WW)          // 1024 floats
#define SLICE_BYTES (SLICE_ELEMS * 4)  // 4096 bytes

template <int N>
__device__ __forceinline__ void wait_asynccnt() {
  asm volatile("s_wait_asynccnt %0" ::"i"(N) : "memory");
}

// Stage one 4KB slice global -> LDS with 8 async B128 ops (512B per op:
// 32 lanes x 16B). GVS addressing: sgpr 64-bit base + vgpr 32-bit offset.
// LDS dest address is the first vgpr, global address is sgpr base + vgpr offset; no imm offset.
__device__ __forceinline__ void async_copy_slice_to_lds(unsigned ldsBase,
                                                        const float* gbase,
                                                        int lane) {
  unsigned long long gb = (unsigned long long)(uintptr_t)gbase;
#pragma unroll
  for (int j = 0; j < 8; ++j) {
    unsigned byteOff = (unsigned)(lane * 16 + j * 512);
    unsigned ldsAddr = ldsBase + byteOff;
    asm volatile("global_load_async_to_lds_b128 %0, %1, %2"
                 :
                 : "v"(ldsAddr), "v"(byteOff), "s"(gb)
                 : "memory");
  }
}

__global__ __launch_bounds__(WAVES_PER_BLOCK * 32) void drop_block_mask_kernel(
    const float* __restrict__ x, const int* __restrict__ T,
    const int* __restrict__ dropBlockPtr, float* __restrict__ out,
    int nSlices) {
  __shared__ float smem[WAVES_PER_BLOCK * 2 * SLICE_ELEMS];  // 32 KB

  const int lane = threadIdx.x & 31;
  const int wave = threadIdx.x >> 5;
  const int half = dropBlockPtr[0] >> 1;  // floor(drop_block/2), drop_block>0

  float* buf0 = &smem[wave * 2 * SLICE_ELEMS];
  float* buf1 = buf0 + SLICE_ELEMS;
  // Generic LDS pointer: low 32 bits == workgroup-relative LDS byte offset.
  const unsigned ldsBase0 = (unsigned)(uintptr_t)(void*)buf0;
  const unsigned ldsBase1 = (unsigned)(uintptr_t)(void*)buf1;

  const int s0 = (blockIdx.x * WAVES_PER_BLOCK + wave) * SLICES_PER_WAVE;
  if (s0 >= nSlices) return;

  // Prologue: stage first slice.
  async_copy_slice_to_lds(ldsBase0, x + (size_t)s0 * SLICE_ELEMS, lane);

#pragma unroll
  for (int i = 0; i < SLICES_PER_WAVE; ++i) {
    const int s = s0 + i;
    if (s >= nSlices) break;

    if (i + 1 < SLICES_PER_WAVE) {
      // Prefetch next slice into the other buffer (clamped addr for safety),
      // then wait until only those 8 new ops remain outstanding -> slice i
      // is resident (async loads complete in order).
      const int sn = (s + 1 < nSlices) ? (s + 1) : (nSlices - 1);
      async_copy_slice_to_lds((i & 1) ? ldsBase0 : ldsBase1,
                              x + (size_t)sn * SLICE_ELEMS, lane);
      wait_asynccnt<8>();
    } else {
      wait_asynccnt<0>();
    }

    const float* buf = (i & 1) ? buf1 : buf0;
    const int sel = T[s];

    // Pass 1: each lane pulls its 32 values (8 x ds_load_b128) and does a
    // lane-local argmax. Lane L covers flat indices e = j*128 + L*4 .. +3,
    // i.e. row r = e>>5, cols (e&31)..(e&31)+3. Strict '>' keeps the first
    // occurrence (matches jnp.argmax).
    float4 vals[8];
    float bestVal = -3.402823466e+38f;
    int bestIdx = 0;
#pragma unroll
    for (int j = 0; j < 8; ++j) {
      const int e = j * 128 + lane * 4;
      vals[j] = *reinterpret_cast<const float4*>(buf + e);
      const float vv[4] = {vals[j].x, vals[j].y, vals[j].z, vals[j].w};
#pragma unroll
      for (int k = 0; k < 4; ++k) {
        if (vv[k] > bestVal) {
          bestVal = vv[k];
          bestIdx = e + k;
        }
      }
    }

    // Wave32 butterfly argmax reduction (first-occurrence tie-break).
#pragma unroll
    for (int m = 16; m >= 1; m >>= 1) {
      const float ov = __shfl_xor(bestVal, m, 32);
      const int oi = __shfl_xor(bestIdx, m, 32);
      if (ov > bestVal || (ov == bestVal && oi < bestIdx)) {
        bestVal = ov;
        bestIdx = oi;
      }
    }

    const int mh = bestIdx >> 5;
    const int mw = bestIdx & (WW - 1);
    const int h1 = max(mh - half, 0);
    const int h2 = min(mh + half, HH - 1);
    const int w1 = max(mw - half, 0);
    const int w2 = min(mw + half, WW - 1);
    const int boxArea = (h2 - h1 + 1) * (w2 - w1 + 1);
    const float lam = (float)SLICE_ELEMS / (float)(SLICE_ELEMS - boxArea);

    // Pass 2: mask + rescale from registers, B128 stores.
    float* o = out + (size_t)s * SLICE_ELEMS;
#pragma unroll
    for (int j = 0; j < 8; ++j) {
      const int e = j * 128 + lane * 4;
      const int r = e >> 5;
      const int c0 = e & (WW - 1);
      const bool inRow = (r >= h1) && (r <= h2);
      float vv[4] = {vals[j].x, vals[j].y, vals[j].z, vals[j].w};
      float ov[4];
#pragma unroll
      for (int k = 0; k < 4; ++k) {
        const int c = c0 + k;
        const bool inBox = inRow && (c >= w1) && (c <= w2);
        const float masked = inBox ? 0.0f : vv[k] * lam;
        ov[k] = sel ? masked : vv[k];
      }
      float4 ovec = make_float4(ov[0], ov[1], ov[2], ov[3]);
      *reinterpret_cast<float4*>(o + e) = ovec;
    }
  }
}

extern "C" void kernel_launch(void* const* d_in, const int* in_sizes, int n_in,
                              void* d_out, int out_size, void* d_ws,
                              size_t ws_size, hipStream_t stream) {
  const float* x = (const float*)d_in[0];
  const int* T = (const int*)d_in[1];
  const int* dropBlock = (const int*)d_in[2];
  float* out = (float*)d_out;

  const int nSlices = in_sizes[0] / SLICE_ELEMS;  // 128*256 = 32768
  const int slicesPerBlock = WAVES_PER_BLOCK * SLICES_PER_WAVE;  // 16
  const int grid = (nSlices + slicesPerBlock - 1) / slicesPerBlock;

  drop_block_mask_kernel<<<grid, WAVES_PER_BLOCK * 32, 0, stream>>>(
      x, T, dropBlock, out, nSlices);
}